// SimpleHGN_24429773980172
// MI455X (gfx1250) — compile-verified
//
#include <hip/hip_runtime.h>
#include <hip/hip_bf16.h>

// ---------------- problem constants (from reference) ----------------
#define NN   50000
#define EE   500000
#define INF_ 256
#define HID_ 64
#define H1_  4
#define NC_  64
#define ED_  32
#define TT_  8
#define SLOPE_ 0.2f

typedef float v2f __attribute__((ext_vector_type(2)));
typedef float v8f __attribute__((ext_vector_type(8)));

// ---------------- float <-> orderable-uint for atomic max ----------------
__device__ __forceinline__ unsigned f_enc(float f) {
  unsigned u = __float_as_uint(f);
  return (u & 0x80000000u) ? ~u : (u | 0x80000000u);
}
__device__ __forceinline__ float f_dec(unsigned u) {
  return __uint_as_float((u & 0x80000000u) ? (u & 0x7FFFFFFFu) : ~u);
}

// LDS byte offset of a __shared__ object: flat LDS aperture keeps the
// wave-relative offset in addr[31:0] (ISA 10.2), so truncation is exact.
__device__ __forceinline__ unsigned lds_off(const void* p) {
  return (unsigned)(uintptr_t)p;
}

// async copy 16B global -> LDS (ASYNCcnt-tracked, no VGPR data movement)
__device__ __forceinline__ void async_b128(unsigned lds_byte, const void* gaddr) {
  asm volatile("global_load_async_to_lds_b128 %0, %1, off"
               :: "v"(lds_byte), "v"((unsigned long long)(uintptr_t)gaddr)
               : "memory");
}
__device__ __forceinline__ void wait_async0() {
  asm volatile("s_wait_asynccnt 0x0" ::: "memory");
}

// ---------------- WMMA GEMM: C[M,Ncol] = A[M,K] @ B[K,Ncol] (+bias) ----------------
// block = 128 threads (4 waves) -> 64x16 C tile. K staged through LDS in
// 64-wide chunks via global_load_async_to_lds_b128. Requires K%64==0, Ncol%16==0.
#define KC    64
#define APAD  68   // 64 + 4: lanes 0-15 read rows 68 floats apart -> banks 4*l, conflict-free

__global__ void __launch_bounds__(128)
hgn_gemm_wmma(const float* __restrict__ A, const float* __restrict__ B,
              const float* __restrict__ bias, float* __restrict__ C,
              int M, int K, int Ncol) {
  __shared__ __align__(16) float As[64 * APAD];   // 17408 B
  __shared__ __align__(16) float Bs[KC * 16];     //  4096 B

  const int tid  = threadIdx.x;
  const int lane = tid & 31;
  const int w    = tid >> 5;          // wave -> m-subtile
  const int half = lane >> 4;         // 0: lanes 0-15, 1: lanes 16-31
  const int l16  = lane & 15;
  const int tn   = blockIdx.x * 16;
  const int tm   = blockIdx.y * 64;

  v8f acc = {};
  for (int kc = 0; kc < K; kc += KC) {
    // ---- stage A tile (64 x 64) : 1024 x b128, 8 per thread ----
    for (int i = tid; i < 1024; i += 128) {
      const int r = i >> 4, c = i & 15;           // row, 16B-chunk
      int rg = tm + r; if (rg >= M) rg = M - 1;   // clamp tail (reads stay in bounds)
      async_b128(lds_off(&As[r * APAD + c * 4]),
                 A + (size_t)rg * K + kc + c * 4);
    }
    // ---- stage B panel (64 x 16) : 256 x b128, 2 per thread ----
    for (int i = tid; i < 256; i += 128) {
      const int r = i >> 2, c = i & 3;
      async_b128(lds_off(&Bs[r * 16 + c * 4]),
                 B + (size_t)(kc + r) * Ncol + tn + c * 4);
    }
    wait_async0();        // this wave's transfers landed in LDS
    __syncthreads();      // everyone's transfers visible

    // ---- 16 WMMAs per chunk, fully unrolled, fed from LDS ----
    const float* arow = &As[(w * 16 + l16) * APAD + half * 2];
#pragma unroll
    for (int kk = 0; kk < KC; kk += 4) {
      const v2f a = *(const v2f*)(arow + kk);               // ds_load_b64
      v2f b;
      b[0] = Bs[(kk + half * 2) * 16 + l16];                // ds_load_b32
      b[1] = Bs[(kk + half * 2 + 1) * 16 + l16];
      acc = __builtin_amdgcn_wmma_f32_16x16x4_f32(
          false, a, false, b, (short)0, acc, false, false);
    }
    __syncthreads();      // protect LDS from next chunk's staging
  }

  // ---- epilogue: C/D layout -> rows m = base + v + 8*half, col = tn + l16 ----
  const int mbase = tm + w * 16;
  const int n = tn + l16;
  const float bv = bias ? bias[n] : 0.0f;
#pragma unroll
  for (int v = 0; v < 8; ++v) {
    const int m = mbase + v + half * 8;
    if (m < M) C[(size_t)m * Ncol + n] = acc[v] + bv;
  }
}

// ---------------- per-etype attention bias: he[t,h] ----------------
__global__ void hgn_etype_bias(const float* __restrict__ edge_emb,
                               const float* __restrict__ Wr,
                               const float* __restrict__ a_e,
                               float* __restrict__ he, int H) {
  int tid = threadIdx.x;
  if (tid >= TT_ * H) return;
  int t = tid / H, h = tid % H;
  float s = 0.0f;
  for (int e = 0; e < ED_; ++e) {
    float w = 0.0f;
    const float* wr = Wr + ((size_t)t * ED_ + e) * (size_t)(ED_ * H) + (size_t)h * ED_;
    for (int d = 0; d < ED_; ++d) w += wr[d] * a_e[h * ED_ + d];
    s += edge_emb[t * ED_ + e] * w;
  }
  he[tid] = s;
}

// ---------------- node dots: hl[n,h]=<emb[n,h,:],a_l[h,:]>, same for hr ----------------
__global__ void hgn_node_dots(const float* __restrict__ emb,
                              const float* __restrict__ a_l,
                              const float* __restrict__ a_r,
                              float* __restrict__ hl, float* __restrict__ hr,
                              int n_nodes, int H) {
  const int wave = (int)((blockIdx.x * (size_t)blockDim.x + threadIdx.x) >> 5);
  const int lane = threadIdx.x & 31;
  if (wave >= n_nodes) return;
  const int D = HID_;
  const float* rowp = emb + (size_t)wave * H * D;
  float pl[H1_], pr[H1_];
  for (int h = 0; h < H1_; ++h) { pl[h] = 0.0f; pr[h] = 0.0f; }
  for (int idx = lane; idx < H * D; idx += 32) {
    int h = idx / D;
    float v = rowp[idx];
    pl[h] += v * a_l[idx];
    pr[h] += v * a_r[idx];
  }
  for (int h = 0; h < H; ++h) {
    float sl = pl[h], sr = pr[h];
    for (int m = 16; m >= 1; m >>= 1) {
      sl += __shfl_xor(sl, m, 32);
      sr += __shfl_xor(sr, m, 32);
    }
    if (lane == 0) { hl[(size_t)wave * H + h] = sl; hr[(size_t)wave * H + h] = sr; }
  }
}

// ---------------- edge attention + segment max ----------------
__global__ void hgn_edge_att(const float* __restrict__ hl, const float* __restrict__ hr,
                             const float* __restrict__ he,
                             const int* __restrict__ row, const int* __restrict__ col,
                             const int* __restrict__ etype,
                             float* __restrict__ att, unsigned* __restrict__ segmax,
                             int E, int H) {
  int e = (int)(blockIdx.x * (size_t)blockDim.x + threadIdx.x);
  if (e >= E) return;
  int r = row[e], c = col[e], t = etype[e];
  for (int h = 0; h < H; ++h) {
    float x = hl[(size_t)r * H + h] + hr[(size_t)c * H + h] + he[t * H + h];
    x = (x > 0.0f) ? x : SLOPE_ * x;
    att[(size_t)e * H + h] = x;
    atomicMax(&segmax[(size_t)c * H + h], f_enc(x));
  }
}

// ---------------- exp(att - max) + segment sum ----------------
__global__ void hgn_edge_exp(const int* __restrict__ col,
                             float* __restrict__ att,
                             const unsigned* __restrict__ segmax,
                             float* __restrict__ segsum, int E, int H) {
  int e = (int)(blockIdx.x * (size_t)blockDim.x + threadIdx.x);
  if (e >= E) return;
  int c = col[e];
  for (int h = 0; h < H; ++h) {
    float m = f_dec(segmax[(size_t)c * H + h]);
    float v = __expf(att[(size_t)e * H + h] - m);
    att[(size_t)e * H + h] = v;
    atomicAdd(&segsum[(size_t)c * H + h], v);
  }
}

// ---------------- layer-1 scatter: out[col, d*H1+h] += emb[row, h*64+d]*alpha ----------------
// one wave per edge; atomic writes are contiguous (D-major target layout),
// the strided access is taken on the gather side.
__global__ void hgn_scatter1(const float* __restrict__ emb, const float* __restrict__ att,
                             const float* __restrict__ segsum,
                             const int* __restrict__ row, const int* __restrict__ col,
                             float* __restrict__ out, int E) {
  const int wave = (int)((blockIdx.x * (size_t)blockDim.x + threadIdx.x) >> 5);
  const int lane = threadIdx.x & 31;
  if (wave >= E) return;
  const int r = row[wave], c = col[wave];
  float alpha[H1_];
  for (int h = 0; h < H1_; ++h)
    alpha[h] = att[(size_t)wave * H1_ + h] / segsum[(size_t)c * H1_ + h];
  const float* src = emb + (size_t)r * (H1_ * HID_);
  float* dst = out + (size_t)c * (H1_ * HID_);
  for (int j = lane; j < H1_ * HID_; j += 32) {
    const int d = j >> 2, h = j & 3;            // dst index j == d*H1 + h
    atomicAdd(&dst[j], src[h * HID_ + d] * alpha[h]);
  }
}

// ---------------- layer-2 scatter: out[col,d] += emb2[row,d]*alpha (adds on residual) ----------------
__global__ void hgn_scatter2(const float* __restrict__ emb2, const float* __restrict__ att,
                             const float* __restrict__ segsum,
                             const int* __restrict__ row, const int* __restrict__ col,
                             float* __restrict__ out, int E) {
  const int wave = (int)((blockIdx.x * (size_t)blockDim.x + threadIdx.x) >> 5);
  const int lane = threadIdx.x & 31;
  if (wave >= E) return;
  const int r = row[wave], c = col[wave];
  const float alpha = att[wave] / segsum[c];
  const float* src = emb2 + (size_t)r * NC_;
  float* dst = out + (size_t)c * NC_;
  for (int idx = lane; idx < NC_; idx += 32)
    atomicAdd(&dst[idx], src[idx] * alpha);
}

// ---------------- ELU ----------------
__global__ void hgn_elu(float* __restrict__ x, size_t n) {
  size_t i = blockIdx.x * (size_t)blockDim.x + threadIdx.x;
  if (i >= n) return;
  float v = x[i];
  x[i] = (v > 0.0f) ? v : (__expf(v) - 1.0f);
}

// ======================= host side =======================
extern "C" void kernel_launch(void* const* d_in, const int* in_sizes, int n_in,
                              void* d_out, int out_size, void* d_ws, size_t ws_size,
                              hipStream_t stream) {
  (void)in_sizes; (void)n_in; (void)out_size; (void)ws_size;
  const float* h        = (const float*)d_in[0];
  const int*   row      = (const int*)  d_in[1];
  const int*   col      = (const int*)  d_in[2];
  const int*   etype    = (const int*)  d_in[3];
  const float* eemb1    = (const float*)d_in[4];
  const float* W1       = (const float*)d_in[5];
  const float* Wr1      = (const float*)d_in[6];
  const float* a_l1     = (const float*)d_in[7];
  const float* a_r1     = (const float*)d_in[8];
  const float* a_e1     = (const float*)d_in[9];
  const float* eemb2    = (const float*)d_in[10];
  const float* W2       = (const float*)d_in[11];
  const float* Wr2      = (const float*)d_in[12];
  const float* a_l2     = (const float*)d_in[13];
  const float* a_r2     = (const float*)d_in[14];
  const float* a_e2     = (const float*)d_in[15];
  const float* res_W2   = (const float*)d_in[16];
  const float* res_b2   = (const float*)d_in[17];
  float* out            = (float*)d_out;

  // ----- workspace carve-up -----
  float* ws   = (float*)d_ws;
  float* emb1 = ws;                               // N*256
  float* h1   = emb1 + (size_t)NN * 256;          // N*256 (zeroed each call)
  float* emb2 = h1   + (size_t)NN * 256;          // N*64
  float* hl1  = emb2 + (size_t)NN * 64;           // N*4
  float* hr1  = hl1  + (size_t)NN * 4;            // N*4
  float* att1 = hr1  + (size_t)NN * 4;            // E*4
  float* ssum1= att1 + (size_t)EE * 4;            // N*4 (zeroed)
  unsigned* smax1 = (unsigned*)(ssum1 + (size_t)NN * 4); // N*4 (zeroed)
  float* hl2  = (float*)(smax1 + (size_t)NN * 4); // N
  float* hr2  = hl2  + NN;                        // N
  float* att2 = hr2  + NN;                        // E
  float* ssum2= att2 + EE;                        // N (zeroed)
  unsigned* smax2 = (unsigned*)(ssum2 + NN);      // N (zeroed)
  float* he1  = (float*)(smax2 + NN);             // T*4
  float* he2  = he1 + TT_ * H1_;                  // T

  // ----- zero the accumulators (graph-capturable memset nodes) -----
  hipMemsetAsync(h1,    0, (size_t)NN * 256 * sizeof(float), stream);
  hipMemsetAsync(ssum1, 0, (size_t)NN * 4 * sizeof(float),   stream);
  hipMemsetAsync(smax1, 0, (size_t)NN * 4 * sizeof(unsigned),stream);
  hipMemsetAsync(ssum2, 0, (size_t)NN * sizeof(float),       stream);
  hipMemsetAsync(smax2, 0, (size_t)NN * sizeof(unsigned),    stream);

  // ----- per-etype attention bias terms -----
  hgn_etype_bias<<<1, 64, 0, stream>>>(eemb1, Wr1, a_e1, he1, H1_);
  hgn_etype_bias<<<1, 64, 0, stream>>>(eemb2, Wr2, a_e2, he2, 1);

  const int MB = (NN + 63) / 64;   // 782 m-tiles of 64 rows

  // ----- layer 1 -----
  {
    dim3 g((H1_ * HID_) / 16, MB);
    hgn_gemm_wmma<<<g, 128, 0, stream>>>(h, W1, nullptr, emb1, NN, INF_, H1_ * HID_);
  }
  hgn_node_dots<<<(NN * 32 + 255) / 256, 256, 0, stream>>>(emb1, a_l1, a_r1, hl1, hr1, NN, H1_);
  hgn_edge_att <<<(EE + 255) / 256, 256, 0, stream>>>(hl1, hr1, he1, row, col, etype,
                                                      att1, smax1, EE, H1_);
  hgn_edge_exp <<<(EE + 255) / 256, 256, 0, stream>>>(col, att1, smax1, ssum1, EE, H1_);
  hgn_scatter1 <<<(EE * 32 + 255) / 256, 256, 0, stream>>>(emb1, att1, ssum1, row, col, h1, EE);
  hgn_elu      <<<((size_t)NN * 256 + 255) / 256, 256, 0, stream>>>(h1, (size_t)NN * 256);

  // ----- layer 2 -----
  {
    dim3 g(NC_ / 16, MB);
    hgn_gemm_wmma<<<g, 128, 0, stream>>>(h1, W2, nullptr, emb2, NN, H1_ * HID_, NC_);
    hgn_gemm_wmma<<<g, 128, 0, stream>>>(h1, res_W2, res_b2, out, NN, H1_ * HID_, NC_);
  }
  hgn_node_dots<<<(NN * 32 + 255) / 256, 256, 0, stream>>>(emb2, a_l2, a_r2, hl2, hr2, NN, 1);
  hgn_edge_att <<<(EE + 255) / 256, 256, 0, stream>>>(hl2, hr2, he2, row, col, etype,
                                                      att2, smax2, EE, 1);
  hgn_edge_exp <<<(EE + 255) / 256, 256, 0, stream>>>(col, att2, smax2, ssum2, EE, 1);
  hgn_scatter2 <<<(EE * 32 + 255) / 256, 256, 0, stream>>>(emb2, att2, ssum2, row, col, out, EE);
}